// IntentGCN_79087527788695
// MI455X (gfx1250) — compile-verified
//
#include <hip/hip_runtime.h>

// Problem constants (from reference)
#define NB   32
#define CIN  64
#define COUT 64
#define KA   3
#define TT   300
#define VN   25
#define HH   4
#define DK   16
#define OO   192   // KA*COUT

typedef float v2f __attribute__((ext_vector_type(2)));
typedef float v8f __attribute__((ext_vector_type(8)));

// ---------------- shared-memory layout (floats) ----------------
// FA   : [32][65]  activations after BN1 (rows>=25 zero)          @ 0     (2080)
// FCG  : [64][26]  f_c[c][w] (graph-conv branch result)           @ 2080  (1664)
// S1/O1/S2/O2 : per-channel BN scale/offset                        @ 3744..4000
// Region (time-multiplexed)                                        @ 4000  (10464)
//   QM / XO / CVO  @ RB+0     (CVO uses [32][201] = 6432)
//   KM / FOUT      @ RB+2080  (FOUT = [64][26])
//   VM             @ RB+4160
//   QN / ATT       @ RB+6240  (ATT = [4][32][33] = 4224)
#define SM_TOTAL 14464
#define FA_OFF   0
#define FA_S     65
#define FCG_OFF  2080
#define FCG_S    26
#define S1_OFF   3744
#define O1_OFF   3808
#define S2_OFF   3872
#define O2_OFF   3936
#define RB       4000
#define QM_OFF   (RB + 0)
#define XO_OFF   (RB + 0)
#define CVO_OFF  (RB + 0)
#define CVO_S    201
#define KM_OFF   (RB + 2080)
#define FOUT_OFF (RB + 2080)
#define VM_OFF   (RB + 4160)
#define QN_OFF   (RB + 6240)
#define ATT_OFF  (RB + 6240)
#define ATT_S    33
#define MAT_S    65   // row stride for 64-wide matrices

// ---------------- WMMA f32 16x16x4 helpers (wave32) ----------------
// A[16x4]: lanes 0-15 -> M=lane, vgpr0=K0, vgpr1=K1; lanes 16-31 -> K2,K3
// B[4x16]: lanes 0-15 -> N=lane, vgpr0=K0, vgpr1=K1; lanes 16-31 -> K2,K3
// C/D[16x16]: vgpr r: lanes 0-15 -> (M=r, N=lane); lanes 16-31 -> (M=r+8, N=lane-16)
__device__ __forceinline__ v2f ldA(const float* p, int rs, int kb, int lane) {
  int m  = lane & 15;
  int k0 = kb + ((lane >> 4) << 1);
  v2f r; r.x = p[m * rs + k0]; r.y = p[m * rs + k0 + 1]; return r;
}
__device__ __forceinline__ v2f ldB(const float* p, int ks, int ns, int kb, int n0, int lane) {
  int nn = n0 + (lane & 15);
  int k0 = kb + ((lane >> 4) << 1);
  v2f r; r.x = p[k0 * ks + nn * ns]; r.y = p[(k0 + 1) * ks + nn * ns]; return r;
}
__device__ __forceinline__ v8f wmma4(v2f a, v2f b, v8f c) {
  return __builtin_amdgcn_wmma_f32_16x16x4_f32(false, a, false, b, (short)0, c, false, false);
}
__device__ __forceinline__ void stD(float* p, int rs, v8f d, int lane) {
  int ro  = (lane >> 4) * 8;
  int col = lane & 15;
#pragma unroll
  for (int r = 0; r < 8; ++r) p[(r + ro) * rs + col] = d[r];
}

__global__ __launch_bounds__(128) void intent_gcn_kernel(
    const float* __restrict__ x,      const float* __restrict__ Ag,
    const float* __restrict__ bn1_g,  const float* __restrict__ bn1_b,
    const float* __restrict__ bn1_m,  const float* __restrict__ bn1_v,
    const float* __restrict__ conv_w, const float* __restrict__ conv_b,
    const float* __restrict__ ln_g,   const float* __restrict__ ln_b,
    const float* __restrict__ wq,     const float* __restrict__ wk,
    const float* __restrict__ wv,     const float* __restrict__ fc_w,
    const float* __restrict__ gate_p,
    const float* __restrict__ bn2_g,  const float* __restrict__ bn2_b,
    const float* __restrict__ bn2_m,  const float* __restrict__ bn2_v,
    float* __restrict__ out)
{
  __shared__ float sm[SM_TOTAL];
  const int tid  = threadIdx.x;
  const int lane = tid & 31;
  const int wid  = tid >> 5;
  const int blk  = blockIdx.x;
  const int n    = blk / TT;
  const int t    = blk % TT;

  // ---- BN affine precompute + zero FA ----
  if (tid < 64) {
    int c = tid;
    float s1 = bn1_g[c] * rsqrtf(bn1_v[c] + 1e-5f);
    sm[S1_OFF + c] = s1;
    sm[O1_OFF + c] = bn1_b[c] - bn1_m[c] * s1;
    float s2 = bn2_g[c] * rsqrtf(bn2_v[c] + 1e-5f);
    sm[S2_OFF + c] = s2;
    sm[O2_OFF + c] = bn2_b[c] - bn2_m[c] * s2;
  }
  for (int e = tid; e < 32 * FA_S; e += 128) sm[FA_OFF + e] = 0.0f;
  __syncthreads();

  // ---- load x slice + BN1 -> FA[v][c] (coalesced along v) ----
  const float* xb = x + (size_t)n * CIN * TT * VN + (size_t)t * VN;
  for (int e = tid; e < CIN * VN; e += 128) {
    int c = e / VN, v = e % VN;
    sm[FA_OFF + v * FA_S + c] =
        xb[(size_t)c * TT * VN + v] * sm[S1_OFF + c] + sm[O1_OFF + c];
  }
  __syncthreads();

  // ---- Phase 1: conv GEMM  CVO[v][o] = FA[32x64] @ conv_w^T[64x192] + conv_b ----
  for (int g = wid; g < 24; g += 4) {
    int mt = g / 12, nt = g % 12;
    const float* Ap = &sm[FA_OFF + mt * 16 * FA_S];
    v8f acc = {0, 0, 0, 0, 0, 0, 0, 0};
#pragma unroll
    for (int kb = 0; kb < 64; kb += 4) {
      v2f a = ldA(Ap, FA_S, kb, lane);
      v2f b = ldB(conv_w, 1, 64, kb, nt * 16, lane);  // B[k=c][n=o] = conv_w[o*64+c]
      acc = wmma4(a, b, acc);
    }
    float cb = conv_b[nt * 16 + (lane & 15)];
    int ro = (lane >> 4) * 8, col = nt * 16 + (lane & 15);
#pragma unroll
    for (int r = 0; r < 8; ++r) {
      int v = mt * 16 + r + ro;
      sm[CVO_OFF + v * CVO_S + col] = (v < VN) ? (acc[r] + cb) : 0.0f;
    }
  }
  __syncthreads();

  // ---- Phase 1b: A-contraction  FCG[c][w] = sum_{k,v} CVO[v][k*64+c] * A[k][v][w] ----
  for (int e = tid; e < COUT * VN; e += 128) {
    int c = e / VN, w = e % VN;
    float acc = 0.0f;
    for (int k = 0; k < KA; ++k) {
      const float* Ak = Ag + (k * VN) * VN + w;          // stride VN over v
      const float* Cv = &sm[CVO_OFF + k * 64 + c];       // stride CVO_S over v
#pragma unroll 5
      for (int v = 0; v < VN; ++v) acc += Cv[v * CVO_S] * Ak[v * VN];
    }
    sm[FCG_OFF + c * FCG_S + w] = acc;
  }
  __syncthreads();

  // ---- Phase 2a: zero QN, then LayerNorm(FA) -> QN (rows < 25) ----
  for (int e = tid; e < 32 * MAT_S; e += 128) sm[QN_OFF + e] = 0.0f;
  __syncthreads();
  if (tid < VN) {
    int v = tid;
    const float* fr = &sm[FA_OFF + v * FA_S];
    float mu = 0.0f;
    for (int c = 0; c < 64; ++c) mu += fr[c];
    mu *= (1.0f / 64.0f);
    float var = 0.0f;
    for (int c = 0; c < 64; ++c) { float d = fr[c] - mu; var += d * d; }
    var *= (1.0f / 64.0f);
    float rstd = rsqrtf(var + 1e-6f);
    for (int c = 0; c < 64; ++c)
      sm[QN_OFF + v * MAT_S + c] = (fr[c] - mu) * rstd * ln_g[c] + ln_b[c];
  }
  __syncthreads();

  // ---- Phase 2b: Q/K/V projections (3 GEMMs, 32x64x64) ----
  for (int g = wid; g < 24; g += 4) {
    int mat = g / 8, sub = g % 8, mt = sub / 4, nt = sub % 4;
    const float* Ap = (mat == 0) ? &sm[QN_OFF + mt * 16 * MAT_S]
                                 : &sm[FA_OFF + mt * 16 * FA_S];
    const float* W = (mat == 0) ? wq : (mat == 1) ? wk : wv;
    float* D = &sm[QM_OFF + mat * 2080];
    v8f acc = {0, 0, 0, 0, 0, 0, 0, 0};
#pragma unroll
    for (int kb = 0; kb < 64; kb += 4) {
      v2f a = ldA(Ap, MAT_S, kb, lane);
      v2f b = ldB(W, 64, 1, kb, nt * 16, lane);   // W[c][d] row-major -> coalesced
      acc = wmma4(a, b, acc);
    }
    stD(D + mt * 16 * MAT_S + nt * 16, MAT_S, acc, lane);
  }
  __syncthreads();

  // ---- Phase 2c: attention scores per head (wave == head), 32x32x16, *0.25 ----
  {
    int h = wid;
    const float* Qh = &sm[QM_OFF + h * DK];
    const float* Kh = &sm[KM_OFF + h * DK];
    for (int tile = 0; tile < 4; ++tile) {
      int mt = tile >> 1, nt = tile & 1;
      v8f acc = {0, 0, 0, 0, 0, 0, 0, 0};
#pragma unroll
      for (int kb = 0; kb < DK; kb += 4) {
        v2f a = ldA(Qh + mt * 16 * MAT_S, MAT_S, kb, lane);
        v2f b = ldB(Kh, 1, MAT_S, kb, nt * 16, lane);  // B[k=d][n=w] = KM[w][h*16+d]
        acc = wmma4(a, b, acc);
      }
      float* D = &sm[ATT_OFF + (h * 32 + mt * 16) * ATT_S + nt * 16];
      int ro = (lane >> 4) * 8, col = lane & 15;
#pragma unroll
      for (int r = 0; r < 8; ++r) D[(r + ro) * ATT_S + col] = acc[r] * 0.25f;
    }
  }
  __syncthreads();

  // ---- Phase 2d: softmax over w (one thread per (h,v) row) ----
  if (tid < HH * VN) {
    int h = tid / VN, v = tid % VN;
    float* row = &sm[ATT_OFF + (h * 32 + v) * ATT_S];
    float mx = row[0];
    for (int w = 1; w < VN; ++w) mx = fmaxf(mx, row[w]);
    float s = 0.0f;
    for (int w = 0; w < VN; ++w) { float e = __expf(row[w] - mx); row[w] = e; s += e; }
    float inv = 1.0f / s;
    for (int w = 0; w < VN; ++w) row[w] *= inv;
    for (int w = VN; w < 32; ++w) row[w] = 0.0f;  // zero pad for K-dim of next GEMM
  }
  __syncthreads();

  // ---- Phase 2e: attn @ V per head -> XO[v][h*16+d]  (32x16x32) ----
  {
    int h = wid;
    const float* Ah = &sm[ATT_OFF + h * 32 * ATT_S];
    const float* Vh = &sm[VM_OFF + h * DK];
    for (int mt = 0; mt < 2; ++mt) {
      v8f acc = {0, 0, 0, 0, 0, 0, 0, 0};
#pragma unroll
      for (int kb = 0; kb < 32; kb += 4) {
        v2f a = ldA(Ah + mt * 16 * ATT_S, ATT_S, kb, lane);
        v2f b = ldB(Vh, MAT_S, 1, kb, 0, lane);   // B[k=w][n=d] = VM[w][h*16+d]
        acc = wmma4(a, b, acc);
      }
      stD(&sm[XO_OFF + mt * 16 * MAT_S + h * DK], MAT_S, acc, lane);
    }
  }
  __syncthreads();

  // ---- Phase 2f: output projection + residual + gated combine -> FOUT[c][w] ----
  {
    float gate = gate_p[0];
    for (int g = wid; g < 8; g += 4) {
      int mt = g / 4, nt = g % 4;
      const float* Ap = &sm[XO_OFF + mt * 16 * MAT_S];
      v8f acc = {0, 0, 0, 0, 0, 0, 0, 0};
#pragma unroll
      for (int kb = 0; kb < 64; kb += 4) {
        v2f a = ldA(Ap, MAT_S, kb, lane);
        v2f b = ldB(fc_w, 64, 1, kb, nt * 16, lane);
        acc = wmma4(a, b, acc);
      }
      int ro = (lane >> 4) * 8, cn = nt * 16 + (lane & 15);
#pragma unroll
      for (int r = 0; r < 8; ++r) {
        int v = mt * 16 + r + ro;
        if (v < VN) {
          float fa_res = sm[FA_OFF + v * FA_S + cn];
          float val = ((acc[r] + fa_res) * gate + sm[FCG_OFF + cn * FCG_S + v]) * 0.5f;
          sm[FOUT_OFF + cn * FCG_S + v] = val;
        }
      }
    }
  }
  __syncthreads();

  // ---- epilogue: BN2 + ReLU, coalesced store along v ----
  float* ob = out + (size_t)n * COUT * TT * VN + (size_t)t * VN;
  for (int e = tid; e < COUT * VN; e += 128) {
    int c = e / VN, v = e % VN;
    float val = sm[FOUT_OFF + c * FCG_S + v] * sm[S2_OFF + c] + sm[O2_OFF + c];
    ob[(size_t)c * TT * VN + v] = fmaxf(val, 0.0f);
  }
}

extern "C" void kernel_launch(void* const* d_in, const int* in_sizes, int n_in,
                              void* d_out, int out_size, void* d_ws, size_t ws_size,
                              hipStream_t stream) {
  (void)in_sizes; (void)n_in; (void)out_size; (void)d_ws; (void)ws_size;
  const float* x      = (const float*)d_in[0];
  const float* A      = (const float*)d_in[1];
  const float* bn1_g  = (const float*)d_in[2];
  const float* bn1_b  = (const float*)d_in[3];
  const float* bn1_m  = (const float*)d_in[4];
  const float* bn1_v  = (const float*)d_in[5];
  const float* conv_w = (const float*)d_in[6];
  const float* conv_b = (const float*)d_in[7];
  const float* ln_g   = (const float*)d_in[8];
  const float* ln_b   = (const float*)d_in[9];
  const float* wq     = (const float*)d_in[10];
  const float* wk     = (const float*)d_in[11];
  const float* wv     = (const float*)d_in[12];
  const float* fc_w   = (const float*)d_in[13];
  const float* gate   = (const float*)d_in[14];
  const float* bn2_g  = (const float*)d_in[15];
  const float* bn2_b  = (const float*)d_in[16];
  const float* bn2_m  = (const float*)d_in[17];
  const float* bn2_v  = (const float*)d_in[18];
  float* out = (float*)d_out;

  dim3 grid(NB * TT);
  dim3 block(128);
  hipLaunchKernelGGL(intent_gcn_kernel, grid, block, 0, stream,
                     x, A, bn1_g, bn1_b, bn1_m, bn1_v, conv_w, conv_b,
                     ln_g, ln_b, wq, wk, wv, fc_w, gate,
                     bn2_g, bn2_b, bn2_m, bn2_v, out);
}